// VectorQuantizer_83081847374192
// MI455X (gfx1250) — compile-verified
//
#include <hip/hip_runtime.h>

typedef __attribute__((ext_vector_type(2))) float v2f;
typedef __attribute__((ext_vector_type(8))) float v8f;
typedef __attribute__((ext_vector_type(4))) unsigned int v4ui;
typedef __attribute__((ext_vector_type(8))) int v8si;
typedef __attribute__((ext_vector_type(4))) int v4si;

#define DIM       64
#define TILE_M    128   // points per workgroup
#define TILE_K    128   // codebook entries per LDS chunk
#define LDS_STR   68    // padded row stride (floats): 272B, 16B aligned, bank shift 4/row

// ---------------------------------------------------------------------------
// TDM: DMA one 128x64 f32 tile (row-major, stride 64) from global into LDS,
// inserting 4 DWORDs of padding after every 64 DWORDs (pad_interval=5,
// pad_amount=3) so rows land at the 68-float padded stride.
// Descriptor layout per CDNA5 ISA.
// ---------------------------------------------------------------------------
__device__ __forceinline__ void tdm_load_tile_128x64(const float* gsrc,
                                                     unsigned lds_byte_off) {
    const unsigned long long ga = (unsigned long long)(uintptr_t)gsrc;
    v4ui g0;
    g0[0] = 1u;                                   // count=1 (valid), user mode
    g0[1] = lds_byte_off;                         // lds_addr (bytes)
    g0[2] = (unsigned)(ga & 0xFFFFFFFFull);       // global_addr[31:0]
    g0[3] = (unsigned)((ga >> 32) & 0x01FFFFFFull)// global_addr[56:32]
          | (2u << 30);                           // type=2 ("image")
    v8si g1;
    g1[0] = (int)((2u << 16)      // data_size = 2 -> 4 bytes
                | (1u << 20)      // pad_enable
                | (5u << 22)      // pad_interval: every 64 DWORDs
                | (3u << 25));    // pad_amount:   4 DWORDs
    g1[1] = (int)(64u << 16);     // [47:32] atomic_barrier_addr=0, tensor_dim0.lo=64
    g1[2] = (int)(128u << 16);    // tensor_dim0.hi=0, tensor_dim1.lo=128
    g1[3] = (int)(64u << 16);     // tensor_dim1.hi=0, tile_dim0=64
    g1[4] = 128;                  // tile_dim1=128, tile_dim2=0
    g1[5] = 64;                   // tensor_dim0_stride.lo = 64
    g1[6] = 0;                    // stride0.hi=0, stride1.lo=0
    g1[7] = 0;
    const v4si gz4 = {0, 0, 0, 0};                 // groups 2/3 unused (2-D tile)
    const v8si gz8 = {0, 0, 0, 0, 0, 0, 0, 0};     // trailing group (clang-23 form)
    __builtin_amdgcn_tensor_load_to_lds(g0, g1, gz4, gz4, gz8, 0);
}

__device__ __forceinline__ unsigned lds_off(const void* p) {
    return (unsigned)(unsigned long long)(uintptr_t)p;  // low 32 bits give the LDS byte offset
}

// ---------------------------------------------------------------------------
// Kernel 1: per-point argmin over the codebook using f32 WMMA (16x16x4).
// Score used: e2[n] - 2 * (x . e_n)  (the ||x||^2 term is row-constant).
// Codebook chunks double-buffered via the Tensor Data Mover.
// ---------------------------------------------------------------------------
__global__ __launch_bounds__(256) void vq_argmin_kernel(
    const float* __restrict__ x, const float* __restrict__ embed,
    int* __restrict__ idx_out, int N, int K)
{
    __shared__ float xs[TILE_M * LDS_STR];
    __shared__ float es[2][TILE_K * LDS_STR];
    __shared__ float e2s[2][TILE_K];

    const int tid      = threadIdx.x;
    const int wave     = tid >> 5;     // 0..7
    const int lane     = tid & 31;
    const int l16      = lane & 15;
    const int hi       = lane >> 4;    // 0 = lanes 0-15, 1 = lanes 16-31
    const int kh       = hi * 2;       // K sub-offset for A/B fragments
    const int row_base = blockIdx.x * TILE_M;
    const bool issuer  = (wave == 0);  // wave 0 drives the TDM
    const int nchunks  = K / TILE_K;   // 16

    // ---- TDM: x tile and codebook chunk 0 (in-order on TENSORcnt)
    if (issuer) {
        tdm_load_tile_128x64(x + (size_t)row_base * DIM, lds_off(xs));
        tdm_load_tile_128x64(embed, lds_off(es[0]));
        __builtin_amdgcn_s_wait_tensorcnt(0);
    }
    __syncthreads();   // publish x tile + chunk 0 to all waves

    // ---- A fragments: wave handles rows [wave*16, wave*16+16)
    // 32-bit A 16x4 layout: lanes 0-15 hold (K=0,K=1) of row=lane; lanes 16-31 hold (K=2,K=3)
    v2f afrag[16];
    {
        const float* arow = &xs[(wave * 16 + l16) * LDS_STR + kh];
#pragma unroll
        for (int k = 0; k < 16; ++k) afrag[k] = *(const v2f*)(arow + k * 4);
    }

    float    minval[8];
    unsigned minidx[8];
#pragma unroll
    for (int j = 0; j < 8; ++j) { minval[j] = __builtin_inff(); minidx[j] = 0u; }

    for (int chunk = 0; chunk < nchunks; ++chunk) {
        const int buf = chunk & 1;

        // kick off DMA of the next chunk into the other buffer (background copy)
        if (issuer && (chunk + 1 < nchunks))
            tdm_load_tile_128x64(embed + (size_t)(chunk + 1) * TILE_K * DIM,
                                 lds_off(es[buf ^ 1]));

        // ---- e2 for this chunk
        if (tid < TILE_K) {
            const float4* er = (const float4*)(&es[buf][tid * LDS_STR]);
            float s = 0.f;
#pragma unroll
            for (int i = 0; i < 16; ++i) {
                float4 v = er[i];
                s = fmaf(v.x, v.x, s); s = fmaf(v.y, v.y, s);
                s = fmaf(v.z, v.z, s); s = fmaf(v.w, v.w, s);
            }
            e2s[buf][tid] = s;
        }
        __syncthreads();   // e2s ready

        // ---- 8 column subtiles of 16 codes each; contraction over D=64, 16 wmma steps
#pragma unroll 1
        for (int st = 0; st < 8; ++st) {
            const float* brow   = &es[buf][(st * 16 + l16) * LDS_STR + kh];
            const float  e2lane = e2s[buf][st * 16 + l16];
            v8f c = {0.f, 0.f, 0.f, 0.f, 0.f, 0.f, 0.f, 0.f};
#pragma unroll
            for (int k = 0; k < 16; ++k) {
                v2f b = *(const v2f*)(brow + k * 4);
                c = __builtin_amdgcn_wmma_f32_16x16x4_f32(
                        false, afrag[k], false, b, (short)0, c, false, false);
            }
            const unsigned n = (unsigned)(chunk * TILE_K + st * 16 + l16);
#pragma unroll
            for (int j = 0; j < 8; ++j) {
                float s  = fmaf(-2.0f, c[j], e2lane);
                bool  lt = s < minval[j];
                minval[j] = lt ? s : minval[j];
                minidx[j] = lt ? n : minidx[j];
            }
        }

        // arrival barrier for chunk+1; also proves everyone finished reading
        // es[buf] before the chunk+2 TDM (issued next iteration) overwrites it
        if (issuer && (chunk + 1 < nchunks))
            __builtin_amdgcn_s_wait_tensorcnt(0);
        __syncthreads();
    }

    // ---- reduce across the 16 lanes holding each row (xor stays within halves)
#pragma unroll
    for (int j = 0; j < 8; ++j) {
        float    v  = minval[j];
        unsigned bi = minidx[j];
#pragma unroll
        for (int off = 8; off >= 1; off >>= 1) {
            float    ov = __shfl_xor(v, off, 32);
            unsigned oi = __shfl_xor(bi, off, 32);
            if (ov < v || (ov == v && oi < bi)) { v = ov; bi = oi; }
        }
        if (l16 == 0) {
            // C/D layout: VGPR j -> row j (lanes 0-15) / row j+8 (lanes 16-31)
            const int row = row_base + wave * 16 + j + (hi ? 8 : 0);
            idx_out[row] = (int)bi;
        }
    }
}

// ---------------------------------------------------------------------------
// Kernel 2: gather quantize = embed[idx], out = x + (q - x), per-block loss sum
// 4 threads per point, 64 points per block.
// ---------------------------------------------------------------------------
__global__ __launch_bounds__(256) void vq_gather_kernel(
    const float* __restrict__ x, const float* __restrict__ embed,
    const int* __restrict__ idx, float* __restrict__ out,
    float* __restrict__ partials)
{
    const int tid = threadIdx.x;
    const int p   = blockIdx.x * 64 + (tid >> 2);
    const int seg = (tid & 3) * 16;

    const int code = idx[p];
    const float4* eq = (const float4*)(embed + (size_t)code * DIM + seg);
    const float4* xq = (const float4*)(x + (size_t)p * DIM + seg);
    float4*       oq = (float4*)(out + (size_t)p * DIM + seg);

    float acc = 0.f;
#pragma unroll
    for (int i = 0; i < 4; ++i) {
        float4 q = eq[i], xv = xq[i];
        float4 d, o;
        d.x = q.x - xv.x; d.y = q.y - xv.y; d.z = q.z - xv.z; d.w = q.w - xv.w;
        o.x = xv.x + d.x; o.y = xv.y + d.y; o.z = xv.z + d.z; o.w = xv.w + d.w;
        acc = fmaf(d.x, d.x, acc); acc = fmaf(d.y, d.y, acc);
        acc = fmaf(d.z, d.z, acc); acc = fmaf(d.w, d.w, acc);
        oq[i] = o;
    }

    __shared__ float red[256];
    red[tid] = acc;
    __syncthreads();
#pragma unroll
    for (int s = 128; s >= 1; s >>= 1) {
        if (tid < s) red[tid] += red[tid + s];
        __syncthreads();
    }
    if (tid == 0) partials[blockIdx.x] = red[0];
}

// ---------------------------------------------------------------------------
// Kernel 3: deterministic single-block reduction of per-block partial sums.
// ---------------------------------------------------------------------------
__global__ __launch_bounds__(256) void vq_loss_reduce(
    const float* __restrict__ partials, float* __restrict__ lvq,
    int nparts, float scale)
{
    const int tid = threadIdx.x;
    float s = 0.f;
    for (int i = tid; i < nparts; i += 256) s += partials[i];  // fixed order
    __shared__ float red[256];
    red[tid] = s;
    __syncthreads();
#pragma unroll
    for (int k = 128; k >= 1; k >>= 1) {
        if (tid < k) red[tid] += red[tid + k];
        __syncthreads();
    }
    if (tid == 0) *lvq = red[0] * scale;
}

// ---------------------------------------------------------------------------
extern "C" void kernel_launch(void* const* d_in, const int* in_sizes, int n_in,
                              void* d_out, int out_size, void* d_ws, size_t ws_size,
                              hipStream_t stream) {
    const float* x     = (const float*)d_in[0];
    const float* embed = (const float*)d_in[1];
    const int N = in_sizes[0] / DIM;   // 131072
    const int K = in_sizes[1] / DIM;   // 2048

    int*   idxbuf   = (int*)d_ws;
    float* partials = (float*)((char*)d_ws + (size_t)N * sizeof(int));
    float* out      = (float*)d_out;
    float* lvq      = out + (size_t)N * DIM;

    const int nblocks2 = N / 64;

    vq_argmin_kernel<<<N / TILE_M, 256, 0, stream>>>(x, embed, idxbuf, N, K);
    vq_gather_kernel<<<nblocks2, 256, 0, stream>>>(x, embed, idxbuf, out, partials);
    vq_loss_reduce<<<1, 256, 0, stream>>>(partials, lvq, nblocks2,
                                          1.0f / ((float)N * (float)DIM));
}